// RANSAC_24464133718447
// MI455X (gfx1250) — compile-verified
//
#include <hip/hip_runtime.h>
#include <math.h>

typedef __attribute__((ext_vector_type(2))) float v2f;
typedef __attribute__((ext_vector_type(8))) float v8f;
typedef __attribute__((ext_vector_type(4))) unsigned int u32x4;
typedef __attribute__((ext_vector_type(4))) int i32x4;
typedef __attribute__((ext_vector_type(8))) int i32x8;

#define RANSAC_THRESH 1.0f
#define NEG_BIG (-3.402823466e+38f)
#define CHUNK 1024   // points staged to LDS per step (24 KB for both images)

// ---------------------------------------------------------------------------
// Tensor Data Mover: 1-D contiguous copy of n_dwords DWORDs global -> LDS.
// D# built per CDNA5 ISA 8.3/8.4 (count=1, type=2 image, data_size=4B,
// tile_dim0=n_dwords, dims 1+ unused). Issued once per wave (EXEC ignored).
// ---------------------------------------------------------------------------
__device__ inline void tdm_load_1d(unsigned int lds_off, const void* gptr,
                                   unsigned int n_dwords,
                                   unsigned int tensor_dwords) {
  unsigned long long ga = (unsigned long long)(uintptr_t)gptr;
  u32x4 g0;
  g0.x = 1u;                                   // count=1, user-mode load
  g0.y = lds_off;                              // lds_addr (bytes)
  g0.z = (unsigned int)ga;                     // global_addr[31:0]
  g0.w = (unsigned int)(ga >> 32) | (2u << 30);// global_addr[56:32] | type=2
  i32x8 g1;
  g1[0] = (int)(2u << 16);                     // workgroup_mask=0, data_size=4B
  g1[1] = (int)((tensor_dwords & 0xFFFFu) << 16);          // tensor_dim0[15:0]
  g1[2] = (int)(((tensor_dwords >> 16) & 0xFFFFu) | (1u << 16)); // dim0 hi | tensor_dim1=1
  g1[3] = (int)((n_dwords & 0xFFFFu) << 16);   // tile_dim0
  g1[4] = 0;                                   // tile_dim1/2 unused
  g1[5] = (int)tensor_dwords;                  // tensor_dim0_stride[31:0]
  g1[6] = 0;
  g1[7] = 0;
  i32x4 z4 = {0, 0, 0, 0};
#if __has_include(<hip/amd_detail/amd_gfx1250_TDM.h>)
  i32x8 z8 = {0, 0, 0, 0, 0, 0, 0, 0};
  __builtin_amdgcn_tensor_load_to_lds(g0, g1, z4, z4, z8, 0);
#else
  __builtin_amdgcn_tensor_load_to_lds(g0, g1, z4, z4, 0);
#endif
}

// ---------------------------------------------------------------------------
// Jacobi eigensolver for small symmetric matrices (row-major NDIM x NDIM).
// ---------------------------------------------------------------------------
template<int NDIM>
__device__ inline void jacobi_eig(float* M, float* V, int sweeps) {
  for (int i = 0; i < NDIM * NDIM; ++i) V[i] = 0.0f;
  for (int i = 0; i < NDIM; ++i) V[i * NDIM + i] = 1.0f;
  for (int s = 0; s < sweeps; ++s) {
    for (int p = 0; p < NDIM - 1; ++p) {
      for (int q = p + 1; q < NDIM; ++q) {
        float apq = M[p * NDIM + q];
        if (fabsf(apq) < 1e-30f) continue;
        float app = M[p * NDIM + p];
        float aqq = M[q * NDIM + q];
        float tau = (aqq - app) / (2.0f * apq);
        float t = (tau >= 0.0f ? 1.0f : -1.0f) /
                  (fabsf(tau) + sqrtf(1.0f + tau * tau));
        float c = rsqrtf(1.0f + t * t);
        float sn = t * c;
        for (int k = 0; k < NDIM; ++k) {
          float mpk = M[p * NDIM + k], mqk = M[q * NDIM + k];
          M[p * NDIM + k] = c * mpk - sn * mqk;
          M[q * NDIM + k] = sn * mpk + c * mqk;
        }
        for (int k = 0; k < NDIM; ++k) {
          float mkp = M[k * NDIM + p], mkq = M[k * NDIM + q];
          M[k * NDIM + p] = c * mkp - sn * mkq;
          M[k * NDIM + q] = sn * mkp + c * mkq;
        }
        for (int k = 0; k < NDIM; ++k) {
          float vkp = V[k * NDIM + p], vkq = V[k * NDIM + q];
          V[k * NDIM + p] = c * vkp - sn * vkq;
          V[k * NDIM + q] = sn * vkp + c * vkq;
        }
      }
    }
  }
}

// ---------------------------------------------------------------------------
// 9x9 AtA -> denormalized, rank-2-projected, F22-normalized F (3x3).
// ---------------------------------------------------------------------------
__device__ inline void solve_F(float* M /*81, destroyed*/,
                               float s1, float mx1, float my1,
                               float s2, float mx2, float my2,
                               float* Fout /*9*/) {
  float V[81];
  jacobi_eig<9>(M, V, 12);
  int imin = 0;
  float dmin = M[0];
  for (int i = 1; i < 9; ++i) {
    float d = M[i * 9 + i];
    if (d < dmin) { dmin = d; imin = i; }
  }
  float F0[9];
  for (int i = 0; i < 9; ++i) F0[i] = V[i * 9 + imin];

  float G[9], VG[9];
  for (int i = 0; i < 3; ++i)
    for (int j = 0; j < 3; ++j) {
      float acc = 0.0f;
      for (int k = 0; k < 3; ++k) acc += F0[k * 3 + i] * F0[k * 3 + j];
      G[i * 3 + j] = acc;
    }
  jacobi_eig<3>(G, VG, 10);
  int jmin = 0;
  float gmin = G[0];
  for (int i = 1; i < 3; ++i) {
    float d = G[i * 3 + i];
    if (d < gmin) { gmin = d; jmin = i; }
  }
  float v3[3] = {VG[0 * 3 + jmin], VG[1 * 3 + jmin], VG[2 * 3 + jmin]};
  float Fv[3];
  for (int i = 0; i < 3; ++i)
    Fv[i] = F0[i * 3 + 0] * v3[0] + F0[i * 3 + 1] * v3[1] + F0[i * 3 + 2] * v3[2];
  float F2[9];
  for (int i = 0; i < 3; ++i)
    for (int j = 0; j < 3; ++j)
      F2[i * 3 + j] = F0[i * 3 + j] - Fv[i] * v3[j];

  float B[9];
  for (int j = 0; j < 3; ++j) {
    B[0 * 3 + j] = s2 * F2[0 * 3 + j];
    B[1 * 3 + j] = s2 * F2[1 * 3 + j];
    B[2 * 3 + j] = -s2 * mx2 * F2[0 * 3 + j] - s2 * my2 * F2[1 * 3 + j] + F2[2 * 3 + j];
  }
  float Fd[9];
  for (int i = 0; i < 3; ++i) {
    Fd[i * 3 + 0] = s1 * B[i * 3 + 0];
    Fd[i * 3 + 1] = s1 * B[i * 3 + 1];
    Fd[i * 3 + 2] = -s1 * mx1 * B[i * 3 + 0] - s1 * my1 * B[i * 3 + 1] + B[i * 3 + 2];
  }
  float inv = 1.0f / Fd[8];
  for (int k = 0; k < 9; ++k) Fout[k] = Fd[k] * inv;
}

// ---------------------------------------------------------------------------
// Kernel 1: one thread per hypothesis — 8-point fit from the sampled octet.
// ---------------------------------------------------------------------------
__global__ __launch_bounds__(256)
void ransac_hypothesis_kernel(const float* __restrict__ pts1,
                              const float* __restrict__ pts2,
                              const int* __restrict__ samples,
                              float* __restrict__ Fall,
                              int* __restrict__ uniqF,
                              int iters) {
  int it = blockIdx.x * blockDim.x + threadIdx.x;
  if (it >= iters) return;

  int idx[8], srt[8];
  for (int k = 0; k < 8; ++k) { idx[k] = samples[it * 8 + k]; srt[k] = idx[k]; }
  for (int a = 0; a < 7; ++a)
    for (int b = 0; b < 7 - a; ++b)
      if (srt[b] > srt[b + 1]) { int t = srt[b]; srt[b] = srt[b + 1]; srt[b + 1] = t; }
  int u = 1;
  for (int k = 0; k < 7; ++k) if (srt[k] == srt[k + 1]) u = 0;
  uniqF[it] = u;

  float x1[8], y1[8], x2[8], y2[8];
  for (int k = 0; k < 8; ++k) {
    x1[k] = pts1[idx[k] * 3 + 0]; y1[k] = pts1[idx[k] * 3 + 1];
    x2[k] = pts2[idx[k] * 3 + 0]; y2[k] = pts2[idx[k] * 3 + 1];
  }
  float mx1 = 0, my1 = 0, mx2 = 0, my2 = 0;
  for (int k = 0; k < 8; ++k) { mx1 += x1[k]; my1 += y1[k]; mx2 += x2[k]; my2 += y2[k]; }
  mx1 *= 0.125f; my1 *= 0.125f; mx2 *= 0.125f; my2 *= 0.125f;
  float md1 = 0, md2 = 0;
  for (int k = 0; k < 8; ++k) {
    float dx = x1[k] - mx1, dy = y1[k] - my1;
    md1 += sqrtf(dx * dx + dy * dy);
    dx = x2[k] - mx2; dy = y2[k] - my2;
    md2 += sqrtf(dx * dx + dy * dy);
  }
  md1 *= 0.125f; md2 *= 0.125f;
  const float SQ2 = 1.41421356237309515f;
  float s1 = SQ2 / (md1 + 1e-8f);
  float s2 = SQ2 / (md2 + 1e-8f);

  float M[81];
  for (int i = 0; i < 81; ++i) M[i] = 0.0f;
  for (int k = 0; k < 8; ++k) {
    float u1 = s1 * (x1[k] - mx1), v1 = s1 * (y1[k] - my1);
    float u2 = s2 * (x2[k] - mx2), v2 = s2 * (y2[k] - my2);
    float a[9] = {u2 * u1, u2 * v1, u2, v2 * u1, v2 * v1, v2, u1, v1, 1.0f};
    for (int r = 0; r < 9; ++r)
      for (int c = 0; c < 9; ++c)
        M[r * 9 + c] += a[r] * a[c];
  }
  float Fres[9];
  solve_F(M, s1, mx1, my1, s2, mx2, my2, Fres);
  for (int k = 0; k < 9; ++k) Fall[it * 9 + k] = Fres[k];
}

// ---------------------------------------------------------------------------
// Kernel 2: Sampson error + accumulation. 8 waves/block share TDM-staged
// LDS point tiles; each wave scores 4 hypotheses via two f32 WMMAs/tile.
// ---------------------------------------------------------------------------
__global__ __launch_bounds__(256)
void sampson_score_kernel(const float* __restrict__ pts1,
                          const float* __restrict__ pts2,
                          const float* __restrict__ Fall,
                          float* __restrict__ cnts,
                          float* __restrict__ ssums,
                          int N, int iters) {
  __shared__ float lds1[3 * CHUNK];
  __shared__ float lds2[3 * CHUNK];
  const int tid = threadIdx.x;
  const int waveInBlk = tid >> 5;
  const int lane = tid & 31;
  const int h0 = (blockIdx.x * 8 + waveInBlk) * 4;   // iters is a multiple of 32
  const int m = lane & 15;
  const bool hi = lane >= 16;
  const int hbase = h0 + (hi ? 2 : 0);

  // A (F rows, for F@p1) and At (F^T rows, for F^T@p2) fragments.
  v2f afrag;  afrag.x = 0.0f;  afrag.y = 0.0f;
  v2f atfrag; atfrag.x = 0.0f; atfrag.y = 0.0f;
  if (m < 6 && h0 < iters) {
    int h = hbase + (m >= 3 ? 1 : 0);
    int r = (m >= 3) ? m - 3 : m;
    const float* Fh = Fall + h * 9;
    if (!hi) {
      afrag.x = Fh[r * 3 + 0]; afrag.y = Fh[r * 3 + 1];
      atfrag.x = Fh[0 * 3 + r]; atfrag.y = Fh[1 * 3 + r];
    } else {
      afrag.x = Fh[r * 3 + 2]; afrag.y = 0.0f;
      atfrag.x = Fh[2 * 3 + r]; atfrag.y = 0.0f;
    }
  }

  const unsigned int lds1_off = (unsigned int)(uintptr_t)(void*)lds1;
  const unsigned int lds2_off = (unsigned int)(uintptr_t)(void*)lds2;

  float cntA = 0.0f, cntB = 0.0f, sumA = 0.0f, sumB = 0.0f;
  const v8f czero = {};

  for (int c0 = 0; c0 < N; c0 += CHUNK) {
    __syncthreads();                 // previous chunk fully consumed
    if (waveInBlk == 0) {            // one TDM issue per instruction per wave
      tdm_load_1d(lds1_off, pts1 + (size_t)c0 * 3, 3 * CHUNK, 3 * (unsigned)N);
      tdm_load_1d(lds2_off, pts2 + (size_t)c0 * 3, 3 * CHUNK, 3 * (unsigned)N);
      __builtin_amdgcn_s_wait_tensorcnt(0);
    }
    __syncthreads();                 // tiles visible to all 8 waves

#pragma unroll 2
    for (int t0 = 0; t0 < CHUNK; t0 += 16) {
      int ln = t0 + m;
      v2f b1, b2;
      if (!hi) {
        b1.x = lds1[ln * 3 + 0]; b1.y = lds1[ln * 3 + 1];   // B[0..1][n]
        b2.x = lds2[ln * 3 + 0]; b2.y = lds2[ln * 3 + 1];
      } else {
        b1.x = lds1[ln * 3 + 2]; b1.y = 0.0f;               // B[2][n]=1, B[3][n]=0
        b2.x = lds2[ln * 3 + 2]; b2.y = 0.0f;
      }
      v8f fx = __builtin_amdgcn_wmma_f32_16x16x4_f32(false, afrag, false, b1,
                                                     (short)0, czero, false, false);
      v8f ft = __builtin_amdgcn_wmma_f32_16x16x4_f32(false, atfrag, false, b2,
                                                     (short)0, czero, false, false);
      float x2 = __shfl(b2.x, m, 32);   // lane m (lo half) holds x2,y2 of point ln
      float y2 = __shfl(b2.y, m, 32);
      {
        float e = x2 * fx[0] + y2 * fx[1] + fx[2];
        float den = fx[0] * fx[0] + fx[1] * fx[1] + ft[0] * ft[0] + ft[1] * ft[1];
        float err = e * e * __builtin_amdgcn_rcpf(den);
        float in = (err <= RANSAC_THRESH) ? 1.0f : 0.0f;
        cntA += in;
        sumA += in * __builtin_amdgcn_sqrtf(err);
      }
      {
        float e = x2 * fx[3] + y2 * fx[4] + fx[5];
        float den = fx[3] * fx[3] + fx[4] * fx[4] + ft[3] * ft[3] + ft[4] * ft[4];
        float err = e * e * __builtin_amdgcn_rcpf(den);
        float in = (err <= RANSAC_THRESH) ? 1.0f : 0.0f;
        cntB += in;
        sumB += in * __builtin_amdgcn_sqrtf(err);
      }
    }
  }

  // Reduce within each 16-lane half (halves hold different hypothesis pairs).
  for (int off = 8; off >= 1; off >>= 1) {
    cntA += __shfl_xor(cntA, off, 32);
    sumA += __shfl_xor(sumA, off, 32);
    cntB += __shfl_xor(cntB, off, 32);
    sumB += __shfl_xor(sumB, off, 32);
  }
  if ((lane & 15) == 0 && h0 < iters) {
    cnts[hbase] = cntA;     ssums[hbase] = sumA;
    cnts[hbase + 1] = cntB; ssums[hbase + 1] = sumB;
  }
}

// ---------------------------------------------------------------------------
// Kernel 3: scoring + argmax (single block).
// ---------------------------------------------------------------------------
__global__ __launch_bounds__(256)
void select_best_kernel(const float* __restrict__ cnts,
                        const float* __restrict__ ssums,
                        const int* __restrict__ uniqF,
                        int iters, int* __restrict__ bestOut) {
  __shared__ float sred[256];
  __shared__ int sidx[256];
  const int tid = threadIdx.x;

  float mlmax = 0.0f;
  for (int i = tid; i < iters; i += 256)
    if (uniqF[i]) mlmax = fmaxf(mlmax, cnts[i]);
  sred[tid] = mlmax;
  __syncthreads();
  for (int s = 128; s > 0; s >>= 1) {
    if (tid < s) sred[tid] = fmaxf(sred[tid], sred[tid + s]);
    __syncthreads();
  }
  const float mlMax = sred[0];
  __syncthreads();

  float mmax = NEG_BIG;
  for (int i = tid; i < iters; i += 256)
    if (uniqF[i]) {
      float means = (ssums[i] + 1e-8f) / (cnts[i] + 1e-8f);
      mmax = fmaxf(mmax, means);
    }
  sred[tid] = mmax;
  __syncthreads();
  for (int s = 128; s > 0; s >>= 1) {
    if (tid < s) sred[tid] = fmaxf(sred[tid], sred[tid + s]);
    __syncthreads();
  }
  const float meansMax = sred[0];
  __syncthreads();

  float best = NEG_BIG;
  int bi = 0x7fffffff;
  for (int i = tid; i < iters; i += 256) {
    float means = (ssums[i] + 1e-8f) / (cnts[i] + 1e-8f);
    float score = 0.5f * (cnts[i] / mlMax) + 0.5f * (1.0f - means / meansMax);
    float v = uniqF[i] ? score : NEG_BIG;
    if (v > best || (v == best && i < bi)) { best = v; bi = i; }
  }
  sred[tid] = best;
  sidx[tid] = bi;
  __syncthreads();
  for (int s = 128; s > 0; s >>= 1) {
    if (tid < s) {
      if (sred[tid + s] > sred[tid] ||
          (sred[tid + s] == sred[tid] && sidx[tid + s] < sidx[tid])) {
        sred[tid] = sred[tid + s];
        sidx[tid] = sidx[tid + s];
      }
    }
    __syncthreads();
  }
  if (tid == 0) bestOut[0] = sidx[0];
}

// ---------------------------------------------------------------------------
// Kernel 4: final weighted 8-point over all N points with the best mask.
// ---------------------------------------------------------------------------
__global__ __launch_bounds__(256)
void final_fit_kernel(const float* __restrict__ pts1,
                      const float* __restrict__ pts2,
                      const float* __restrict__ Fall,
                      const int* __restrict__ bestIdx,
                      float* __restrict__ out, int N) {
  __shared__ float sF[9];
  __shared__ float sAcc[8];
  __shared__ float sAtA[81];
  const int tid = threadIdx.x;
  const int nth = blockDim.x;

  if (tid < 9) sF[tid] = Fall[bestIdx[0] * 9 + tid];
  if (tid < 8) sAcc[tid] = 0.0f;
  for (int i = tid; i < 81; i += nth) sAtA[i] = 0.0f;
  __syncthreads();

  const float f00 = sF[0], f01 = sF[1], f02 = sF[2];
  const float f10 = sF[3], f11 = sF[4], f12 = sF[5];
  const float f20 = sF[6], f21 = sF[7], f22 = sF[8];

  float sw = 0, sx1 = 0, sy1 = 0, sx2 = 0, sy2 = 0;
  for (int n = tid; n < N; n += nth) {
    float x1 = pts1[n * 3 + 0], y1 = pts1[n * 3 + 1];
    float x2 = pts2[n * 3 + 0], y2 = pts2[n * 3 + 1];
    float fx0 = f00 * x1 + f01 * y1 + f02;
    float fx1v = f10 * x1 + f11 * y1 + f12;
    float fx2v = f20 * x1 + f21 * y1 + f22;
    float ft0 = f00 * x2 + f10 * y2 + f20;
    float ft1 = f01 * x2 + f11 * y2 + f21;
    float e = x2 * fx0 + y2 * fx1v + fx2v;
    float den = fx0 * fx0 + fx1v * fx1v + ft0 * ft0 + ft1 * ft1;
    float err = e * e / den;
    float w = (err <= RANSAC_THRESH) ? 1.0f : 0.0f;
    out[9 + n] = w;
    sw += w; sx1 += w * x1; sy1 += w * y1; sx2 += w * x2; sy2 += w * y2;
  }
  atomicAdd(&sAcc[0], sw);
  atomicAdd(&sAcc[1], sx1);
  atomicAdd(&sAcc[2], sy1);
  atomicAdd(&sAcc[3], sx2);
  atomicAdd(&sAcc[4], sy2);
  __syncthreads();
  const float SW = sAcc[0];
  const float mx1 = sAcc[1] / SW, my1 = sAcc[2] / SW;
  const float mx2 = sAcc[3] / SW, my2 = sAcc[4] / SW;

  float sd1 = 0, sd2 = 0;
  for (int n = tid; n < N; n += nth) {
    float w = out[9 + n];
    float dx = pts1[n * 3 + 0] - mx1, dy = pts1[n * 3 + 1] - my1;
    sd1 += w * sqrtf(dx * dx + dy * dy);
    dx = pts2[n * 3 + 0] - mx2; dy = pts2[n * 3 + 1] - my2;
    sd2 += w * sqrtf(dx * dx + dy * dy);
  }
  atomicAdd(&sAcc[5], sd1);
  atomicAdd(&sAcc[6], sd2);
  __syncthreads();
  const float SQ2 = 1.41421356237309515f;
  const float s1 = SQ2 / (sAcc[5] / SW + 1e-8f);
  const float s2 = SQ2 / (sAcc[6] / SW + 1e-8f);

  float loc[45];
  for (int k = 0; k < 45; ++k) loc[k] = 0.0f;
  for (int n = tid; n < N; n += nth) {
    float w = out[9 + n];
    if (w > 0.0f) {
      float u1 = s1 * (pts1[n * 3 + 0] - mx1), v1 = s1 * (pts1[n * 3 + 1] - my1);
      float u2 = s2 * (pts2[n * 3 + 0] - mx2), v2 = s2 * (pts2[n * 3 + 1] - my2);
      float a[9] = {u2 * u1, u2 * v1, u2, v2 * u1, v2 * v1, v2, u1, v1, 1.0f};
      int k = 0;
      for (int r = 0; r < 9; ++r)
        for (int c = r; c < 9; ++c)
          loc[k++] += a[r] * a[c];
    }
  }
  {
    int k = 0;
    for (int r = 0; r < 9; ++r)
      for (int c = r; c < 9; ++c)
        atomicAdd(&sAtA[r * 9 + c], loc[k++]);
  }
  __syncthreads();

  if (tid == 0) {
    float M[81];
    for (int r = 0; r < 9; ++r)
      for (int c = 0; c < 9; ++c)
        M[r * 9 + c] = (r <= c) ? sAtA[r * 9 + c] : sAtA[c * 9 + r];
    float Fres[9];
    solve_F(M, s1, mx1, my1, s2, mx2, my2, Fres);
    for (int k = 0; k < 9; ++k) out[k] = Fres[k];
  }
}

// ---------------------------------------------------------------------------
extern "C" void kernel_launch(void* const* d_in, const int* in_sizes, int n_in,
                              void* d_out, int out_size, void* d_ws, size_t ws_size,
                              hipStream_t stream) {
  const float* pts1 = (const float*)d_in[0];
  const float* pts2 = (const float*)d_in[1];
  const int* samples = (const int*)d_in[2];
  const int N = in_sizes[0] / 3;       // 16384
  const int iters = in_sizes[2] / 8;   // 4096 (K == 8)

  float* Fall = (float*)d_ws;                  // iters*9
  float* cnts = Fall + (size_t)iters * 9;      // iters
  float* ssums = cnts + iters;                 // iters
  int* uniqF = (int*)(ssums + iters);          // iters
  int* best = uniqF + iters;                   // 1

  ransac_hypothesis_kernel<<<(iters + 255) / 256, 256, 0, stream>>>(
      pts1, pts2, samples, Fall, uniqF, iters);

  // 8 waves/block, 4 hypotheses/wave -> 32 hypotheses per block
  sampson_score_kernel<<<(iters + 31) / 32, 256, 0, stream>>>(
      pts1, pts2, Fall, cnts, ssums, N, iters);

  select_best_kernel<<<1, 256, 0, stream>>>(cnts, ssums, uniqF, iters, best);

  final_fit_kernel<<<1, 256, 0, stream>>>(pts1, pts2, Fall, best,
                                          (float*)d_out, N);
}